// GeGate_30794915512766
// MI455X (gfx1250) — compile-verified
//
#include <hip/hip_runtime.h>

// Elementwise, HBM-bandwidth-bound: y = x*w + b (w,b broadcast per column),
// then CappedReLU (keep y in (0,1], else 0). fp32 in / fp32 out.
// Strategy: column-stationary w/b in registers, 128-bit non-temporal
// streaming loads/stores for x/out, global_prefetch for the next row.

typedef float v4f __attribute__((ext_vector_type(4)));

__device__ __forceinline__ float capped_relu(float v) {
    return (v > 0.0f && v <= 1.0f) ? v : 0.0f;
}

// Vectorized path: D % 4 == 0. Each thread owns one float4 column slice and
// streams rows [r0, r1).
__global__ void scale_bias_cappedrelu_v4(const v4f* __restrict__ x,
                                         const v4f* __restrict__ w,
                                         const v4f* __restrict__ b,
                                         v4f* __restrict__ out,
                                         int D4, int B, int rows_per_block) {
    const int col = blockIdx.x * blockDim.x + threadIdx.x;
    if (col >= D4) return;

    // w/b loaded once per thread; reused across all rows -> zero repeat traffic.
    const v4f w4 = w[col];
    const v4f b4 = b[col];

    const int r0 = blockIdx.y * rows_per_block;
    int r1 = r0 + rows_per_block;
    if (r1 > B) r1 = B;

    const v4f* __restrict__ xp = x + (size_t)r0 * (size_t)D4 + col;
    v4f* __restrict__ op       = out + (size_t)r0 * (size_t)D4 + col;

    for (int r = r0; r < r1; ++r) {
        // Prefetch next row's cacheline (gfx1250 global_prefetch_b8).
        __builtin_prefetch((const void*)(xp + D4), 0, 0);

        // Streamed data, no reuse: non-temporal B128 load.
        v4f v = __builtin_nontemporal_load(xp);

        v4f y;
        y.x = fmaf(v.x, w4.x, b4.x);
        y.y = fmaf(v.y, w4.y, b4.y);
        y.z = fmaf(v.z, w4.z, b4.z);
        y.w = fmaf(v.w, w4.w, b4.w);

        v4f rres;
        rres.x = capped_relu(y.x);
        rres.y = capped_relu(y.y);
        rres.z = capped_relu(y.z);
        rres.w = capped_relu(y.w);

        // Non-temporal B128 store.
        __builtin_nontemporal_store(rres, op);

        xp += D4;
        op += D4;
    }
}

// Scalar fallback (only used if D % 4 != 0; not hit for D = 4096).
__global__ void scale_bias_cappedrelu_scalar(const float* __restrict__ x,
                                             const float* __restrict__ w,
                                             const float* __restrict__ b,
                                             float* __restrict__ out,
                                             long long n, int D) {
    long long i = (long long)blockIdx.x * blockDim.x + threadIdx.x;
    long long stride = (long long)gridDim.x * blockDim.x;
    for (; i < n; i += stride) {
        int c = (int)(i % D);
        float y = fmaf(x[i], w[c], b[c]);
        out[i] = capped_relu(y);
    }
}

extern "C" void kernel_launch(void* const* d_in, const int* in_sizes, int n_in,
                              void* d_out, int out_size, void* d_ws, size_t ws_size,
                              hipStream_t stream) {
    const float* x = (const float*)d_in[0];
    const float* w = (const float*)d_in[1];
    const float* b = (const float*)d_in[2];
    float* out = (float*)d_out;

    const long long n = (long long)in_sizes[0]; // B*D elements
    const int D = in_sizes[1];                  // 4096
    const int B = (int)(n / D);                 // 8192

    if ((D & 3) == 0) {
        const int D4 = D >> 2;                  // 1024 float4 per row
        const int block = 256;                  // 8 wave32 waves per block
        const int gx = (D4 + block - 1) / block; // 4 blocks across columns

        // Aim for ~4096 total blocks to keep all WGPs streaming.
        int gy = 4096 / gx;
        if (gy < 1) gy = 1;
        if (gy > B) gy = B;
        const int rows_per_block = (B + gy - 1) / gy;

        dim3 grid(gx, gy, 1);
        scale_bias_cappedrelu_v4<<<grid, block, 0, stream>>>(
            (const v4f*)x, (const v4f*)w, (const v4f*)b, (v4f*)out,
            D4, B, rows_per_block);
    } else {
        const int block = 256;
        int blocks = (int)((n + block - 1) / block);
        if (blocks > 4096) blocks = 4096;
        scale_bias_cappedrelu_scalar<<<blocks, block, 0, stream>>>(
            x, w, b, out, n, D);
    }
}